// Rpool_15444702397075
// MI455X (gfx1250) — compile-verified
//
#include <hip/hip_runtime.h>
#include <hip/hip_bf16.h>
#include <stdint.h>

// ---------------------------------------------------------------------------
// R-MAC pooling for x[16, 2048, 48, 64] f32.
//   Stage 1: 21 static region maxes per (b,c) plane  (403 MB stream -> 2.75 MB)
//   Stage 2: per-(b,region) inverse L2 norm over channels
//   Stage 3: per-b aggregate + final L2 normalize  -> out[16, 2048]
// Stage 1 uses the CDNA5 async global->LDS copy path (GLOBAL_LOAD_ASYNC_TO_LDS_B128
// + s_wait_asynccnt) with a double-buffered 8-row pipeline per wave32.
// ---------------------------------------------------------------------------

constexpr int kB = 16;
constexpr int kC = 2048;
constexpr int kH = 48;
constexpr int kW = 64;
constexpr int kR = 21;          // 1 global + 2 + 6 + 12 regions
constexpr float kEps = 1e-6f;

// Region table replicated from _region_coords(48, 64, 3) (trace-time constants).
// r0 = global (48x64). l=1: wl=48, j in {0,16}. l=2: wl=32, i in {0,16}, j in {0,16,32}.
// l=3: wl=24, i in {0,12,24}, j in {0,13,26,40}.
__device__ __constant__ const int c_dummy = 0; // keep constant section non-empty

__global__ __launch_bounds__(32)
void rmac_regmax(const float* __restrict__ x, float* __restrict__ rmax) {
  constexpr int RI[kR] = {0,  0, 0,   0, 0, 0, 16, 16, 16,
                          0, 0, 0, 0, 12, 12, 12, 12, 24, 24, 24, 24};
  constexpr int RJ[kR] = {0,  0, 16,  0, 16, 32, 0, 16, 32,
                          0, 13, 26, 40, 0, 13, 26, 40, 0, 13, 26, 40};
  constexpr int RH[kR] = {48, 48, 48, 32, 32, 32, 32, 32, 32,
                          24, 24, 24, 24, 24, 24, 24, 24, 24, 24, 24, 24};
  constexpr int RW[kR] = {64, 48, 48, 32, 32, 32, 32, 32, 32,
                          24, 24, 24, 24, 24, 24, 24, 24, 24, 24, 24, 24};

  const int plane = blockIdx.x;              // b*C + c
  const int lane  = threadIdx.x;             // 0..31 (wave32)
  const float* __restrict__ px = x + (size_t)plane * (kH * kW);
  const int w0 = 2 * lane;
  const int w1 = 2 * lane + 1;

  // Per-lane column-band membership per region (compile-time folded per r).
  bool m0[kR], m1[kR];
#pragma unroll
  for (int r = 1; r < kR; ++r) {
    m0[r] = (w0 >= RJ[r]) && (w0 < RJ[r] + RW[r]);
    m1[r] = (w1 >= RJ[r]) && (w1 < RJ[r] + RW[r]);
  }

  float acc[kR];
#pragma unroll
  for (int r = 0; r < kR; ++r) acc[r] = -__builtin_inff();

  // Double-buffered LDS staging: 2 chunks x 8 rows x 64 cols x 4 B = 4 KB.
  __shared__ float lds[2][8 * kW];
  const uint32_t ldsbase = (uint32_t)(uintptr_t)(&lds[0][0]);
  constexpr int kChunkBytes = 8 * kW * 4;    // 2048 B per chunk
  constexpr int kNChunks    = kH / 8;        // 6 chunks

  // Issue one chunk: 32 lanes x 4 x B128 = 2048 B async copy global -> LDS (GVS mode).
  auto issue_chunk = [&](int ch) {
    const uint32_t lbase = ldsbase + (uint32_t)((ch & 1) * kChunkBytes);
#pragma unroll
    for (int k = 0; k < 4; ++k) {
      const uint32_t loff = lbase + (uint32_t)(k * 512 + lane * 16);
      const uint32_t goff = (uint32_t)(ch * kChunkBytes + k * 512 + lane * 16);
      asm volatile("global_load_async_to_lds_b128 %0, %1, %2 offset:0"
                   :
                   : "v"(loff), "v"(goff), "s"(px)
                   : "memory");
    }
  };

  issue_chunk(0);
#pragma unroll
  for (int ch = 0; ch < kNChunks; ++ch) {
    if (ch + 1 < kNChunks) {
      issue_chunk(ch + 1);
      // 4 outstanding allowed = next chunk still in flight; chunk `ch` complete
      // (async loads complete in order).
      asm volatile("s_wait_asynccnt 4" ::: "memory");
    } else {
      asm volatile("s_wait_asynccnt 0" ::: "memory");
    }
    const float* __restrict__ buf = &lds[ch & 1][0];
#pragma unroll
    for (int r8 = 0; r8 < 8; ++r8) {
      const int h = ch * 8 + r8;               // compile-time after unroll
      const float2 v = *reinterpret_cast<const float2*>(&buf[r8 * kW + 2 * lane]);
      acc[0] = fmaxf(acc[0], fmaxf(v.x, v.y)); // global region: all columns
#pragma unroll
      for (int r = 1; r < kR; ++r) {
        if (h >= RI[r] && h < RI[r] + RH[r]) { // folded at compile time
          const float c0 = m0[r] ? v.x : -__builtin_inff();
          const float c1 = m1[r] ? v.y : -__builtin_inff();
          acc[r] = fmaxf(acc[r], fmaxf(c0, c1));
        }
      }
    }
  }

  // Cross-lane max butterfly; lane r keeps region r's result for a coalesced store.
  float outv = 0.0f;
#pragma unroll
  for (int r = 0; r < kR; ++r) {
    float v = acc[r];
#pragma unroll
    for (int s = 16; s > 0; s >>= 1) v = fmaxf(v, __shfl_xor(v, s, 32));
    if (lane == r) outv = v;
  }
  if (lane < kR) rmax[(size_t)plane * kR + lane] = outv;
}

// ---------------------------------------------------------------------------
// Stage 2: invnrm[b][r] = 1 / (||rmax[b, :, r]||_2 + eps)
// ---------------------------------------------------------------------------
__global__ __launch_bounds__(256)
void rmac_norms(const float* __restrict__ rmax, float* __restrict__ invnrm) {
  const int b = blockIdx.x / kR;
  const int r = blockIdx.x % kR;
  __shared__ float sred[256];
  float s = 0.0f;
  for (int c = threadIdx.x; c < kC; c += 256) {
    const float v = rmax[((size_t)b * kC + c) * kR + r];
    s += v * v;
  }
  sred[threadIdx.x] = s;
  __syncthreads();
#pragma unroll
  for (int k = 128; k > 0; k >>= 1) {
    if (threadIdx.x < k) sred[threadIdx.x] += sred[threadIdx.x + k];
    __syncthreads();
  }
  if (threadIdx.x == 0) invnrm[b * kR + r] = 1.0f / (sqrtf(sred[0]) + kEps);
}

// ---------------------------------------------------------------------------
// Stage 3 (fused): o[b][c] = sum_r rmax[b][c][r] * invnrm[b][r];
//                  out[b][c] = o / (||o[b,:]||_2 + eps)
// One block per batch element.
// ---------------------------------------------------------------------------
__global__ __launch_bounds__(256)
void rmac_agg(const float* __restrict__ rmax, const float* __restrict__ invnrm,
              float* __restrict__ out) {
  const int b = blockIdx.x;
  __shared__ float so[kC];      // 8 KB
  __shared__ float sred[256];

  float inr[kR];
#pragma unroll
  for (int r = 0; r < kR; ++r) inr[r] = invnrm[b * kR + r];

  float ssq = 0.0f;
  for (int c = threadIdx.x; c < kC; c += 256) {
    const float* __restrict__ p = &rmax[((size_t)b * kC + c) * kR];
    float o = 0.0f;
#pragma unroll
    for (int r = 0; r < kR; ++r) o += p[r] * inr[r];
    so[c] = o;
    ssq += o * o;
  }
  sred[threadIdx.x] = ssq;
  __syncthreads();
#pragma unroll
  for (int k = 128; k > 0; k >>= 1) {
    if (threadIdx.x < k) sred[threadIdx.x] += sred[threadIdx.x + k];
    __syncthreads();
  }
  const float scale = 1.0f / (sqrtf(sred[0]) + kEps);
  for (int c = threadIdx.x; c < kC; c += 256)
    out[(size_t)b * kC + c] = so[c] * scale;
}

// ---------------------------------------------------------------------------
extern "C" void kernel_launch(void* const* d_in, const int* in_sizes, int n_in,
                              void* d_out, int out_size, void* d_ws, size_t ws_size,
                              hipStream_t stream) {
  (void)in_sizes; (void)n_in; (void)out_size; (void)ws_size;
  const float* x = (const float*)d_in[0];
  float* out = (float*)d_out;

  float* rmax   = (float*)d_ws;                         // B*C*21 floats (~2.75 MB)
  float* invnrm = rmax + (size_t)kB * kC * kR;          // B*21 floats

  rmac_regmax<<<kB * kC, 32, 0, stream>>>(x, rmax);
  rmac_norms<<<kB * kR, 256, 0, stream>>>(rmax, invnrm);
  rmac_agg  <<<kB, 256, 0, stream>>>(rmax, invnrm, out);
}